// Encoder_12197707121061
// MI455X (gfx1250) — compile-verified
//
#include <hip/hip_runtime.h>

typedef __attribute__((ext_vector_type(2))) float v2f;
typedef __attribute__((ext_vector_type(8))) float v8f;

constexpr int Bc = 2, Sc = 2048, Dc = 512, Hc = 8, Fc = 2048, Lc = 4, HDc = 64;
constexpr int BS = Bc * Sc;              // 4096 token rows
constexpr size_t BSD = (size_t)BS * Dc;  // elements in one [B,S,D] buffer
constexpr float EPSc = 1e-5f;
constexpr int KC = 32;                   // GEMM K-chunk staged in LDS

// ---- wave32 16-lane group reductions (xor masks 1/2/4/8 stay within each half) ----
__device__ inline float grpmax16(float v) {
  v = fmaxf(v, __shfl_xor(v, 1));
  v = fmaxf(v, __shfl_xor(v, 2));
  v = fmaxf(v, __shfl_xor(v, 4));
  v = fmaxf(v, __shfl_xor(v, 8));
  return v;
}
__device__ inline float grpsum16(float v) {
  v += __shfl_xor(v, 1);
  v += __shfl_xor(v, 2);
  v += __shfl_xor(v, 4);
  v += __shfl_xor(v, 8);
  return v;
}

// ---------------- embedding + sinusoidal positional encoding ----------------
__global__ __launch_bounds__(256) void embed_pe_kernel(const int* __restrict__ tokens,
                                                       const float* __restrict__ emb,
                                                       float* __restrict__ x) {
  const int row = blockIdx.x;  // 0..B*S-1
  const int s = row % Sc;
  const int tok = tokens[row];
  const float c = -9.210340371976184f / (float)Dc;  // -ln(10000)/D
  for (int d = threadIdx.x; d < Dc; d += 256) {
    const int i2 = d & ~1;
    const float ang = (float)s * __expf((float)i2 * c);
    const float pe = (d & 1) ? __cosf(ang) : __sinf(ang);
    x[(size_t)row * Dc + d] = emb[(size_t)tok * Dc + d] + pe;
  }
}

// ---------------- fp32 WMMA GEMM with async-to-LDS staging ----------------
// Block = 8 waves computing a 64x128 output tile (each wave a 64x16 strip).
// Per 32-wide K chunk: stage A[64][32] (8KB) and W[32][128] (16KB) into LDS via
// GLOBAL_LOAD_ASYNC_TO_LDS_B128 (ASYNCcnt), then 8 WMMA k-steps from LDS.
__global__ __launch_bounds__(256)
void gemm_wmma_f32(const float* __restrict__ A, const float* __restrict__ W,
                   const float* __restrict__ bias, float* __restrict__ C,
                   int M, int N, int K, int relu) {
  __shared__ float sA[64 * KC];    // [64 rows][KC] row-major
  __shared__ float sW[KC * 128];   // [KC rows][128 cols] row-major
  const int tid = threadIdx.x;
  const int wid = tid >> 5;
  const int lane = tid & 31;
  const int half = lane >> 4, l16 = lane & 15;
  const int ntiles = N >> 4;
  const int gw0 = blockIdx.x * 8;
  const int mt = gw0 / ntiles;          // block-uniform (8 divides ntiles)
  const int ntB = gw0 % ntiles;         // first n-tile of this block
  const int m0 = mt << 6;
  const int n0b = ntB << 4;             // block column base (covers 128 cols)
  const int nLoc = (wid << 4) + l16;    // 0..127 : this lane's column in block
  (void)M;
  const unsigned sAbase = (unsigned)(size_t)sA;  // LDS byte offset = addr[31:0]
  const unsigned sWbase = (unsigned)(size_t)sW;
  v8f acc[4] = {};
  for (int kc = 0; kc < K; kc += KC) {
    __syncthreads();  // protect LDS against overwrite while prior chunk in use
    // stage A chunk: 512 x 16B segments across 256 threads
#pragma unroll
    for (int j = 0; j < 2; ++j) {
      const int seg = tid + j * 256;           // 0..511
      const int row = seg >> 3, c4 = (seg & 7) << 2;
      const unsigned long long ga =
          (unsigned long long)(A + (size_t)(m0 + row) * K + kc + c4);
      const unsigned la = sAbase + (unsigned)((row * KC + c4) << 2);
      asm volatile("global_load_async_to_lds_b128 %0, %1, off"
                   :: "v"(la), "v"(ga) : "memory");
    }
    // stage W chunk: 1024 x 16B segments across 256 threads
#pragma unroll
    for (int j = 0; j < 4; ++j) {
      const int seg = tid + j * 256;           // 0..1023
      const int r = seg >> 5, c4 = (seg & 31) << 2;
      const unsigned long long ga =
          (unsigned long long)(W + (size_t)(kc + r) * N + n0b + c4);
      const unsigned la = sWbase + (unsigned)((r * 128 + c4) << 2);
      asm volatile("global_load_async_to_lds_b128 %0, %1, off"
                   :: "v"(la), "v"(ga) : "memory");
    }
    asm volatile("s_wait_asynccnt 0x0" ::: "memory");  // own async loads done
    __syncthreads();                                    // all waves' loads done
    // compute 8 k-steps from LDS
#pragma unroll
    for (int k0 = 0; k0 < KC; k0 += 4) {
      const int kk = k0 + 2 * half;  // this lane-half holds K = kk, kk+1
      v2f bf;
      bf.x = sW[kk * 128 + nLoc];
      bf.y = sW[(kk + 1) * 128 + nLoc];
#pragma unroll
      for (int t = 0; t < 4; ++t) {
        const v2f af = *(const v2f*)(sA + (t * 16 + l16) * KC + kk);
        acc[t] = __builtin_amdgcn_wmma_f32_16x16x4_f32(false, af, false, bf,
                                                       (short)0, acc[t], false, false);
      }
    }
  }
  const float bv = bias ? bias[n0b + nLoc] : 0.0f;
#pragma unroll
  for (int t = 0; t < 4; ++t) {
#pragma unroll
    for (int r = 0; r < 8; ++r) {  // C/D row = r + 8*half, col = nLoc
      const int m = m0 + t * 16 + r + 8 * half;
      float v = acc[t][r] + bv;
      if (relu) v = fmaxf(v, 0.0f);
      C[(size_t)m * N + n0b + nLoc] = v;
    }
  }
}

// ---------------- flash attention, fp32 WMMA, one wave per 16-query tile ----------------
__global__ __launch_bounds__(256)
void flash_attn_wmma(const float* __restrict__ Q, const float* __restrict__ Km,
                     const float* __restrict__ Vm, float* __restrict__ O) {
  __shared__ float pbuf[8][16][17];  // per-wave P tile, padded stride
  const int wid = threadIdx.x >> 5, lane = threadIdx.x & 31;
  const int half = lane >> 4, l16 = lane & 15;
  const int gw = blockIdx.x * 8 + wid;
  const int qtiles = Sc / 16;
  const int qt = gw % qtiles, bh = gw / qtiles;
  const int b = bh / Hc, h = bh % Hc;
  const float* qp = Q + (size_t)b * Sc * Dc + h * HDc;
  const float* kp = Km + (size_t)b * Sc * Dc + h * HDc;
  const float* vp = Vm + (size_t)b * Sc * Dc + h * HDc;
  const int q0 = qt * 16;
  v8f oacc[4] = {};
  float Mrow[8], Lrow[8];
#pragma unroll
  for (int r = 0; r < 8; ++r) { Mrow[r] = -3.0e38f; Lrow[r] = 0.0f; }

  for (int kt = 0; kt < Sc; kt += 16) {
    // scores tile S(16x16) = Qtile(16x64) @ K^T(64x16)
    v8f sc = {};
#pragma unroll
    for (int d0 = 0; d0 < HDc; d0 += 4) {
      const int dd = d0 + 2 * half;
      const v2f a = *(const v2f*)(qp + (size_t)(q0 + l16) * Dc + dd);
      const v2f bb = *(const v2f*)(kp + (size_t)(kt + l16) * Dc + dd);  // B[d][n]=K[n][d]
      sc = __builtin_amdgcn_wmma_f32_16x16x4_f32(false, a, false, bb,
                                                 (short)0, sc, false, false);
    }
    // online softmax over this key tile
#pragma unroll
    for (int r = 0; r < 8; ++r) {
      const float v = sc[r] * 0.125f;  // 1/sqrt(64)
      const float tmax = grpmax16(v);
      const float Mn = fmaxf(Mrow[r], tmax);
      const float alpha = __expf(Mrow[r] - Mn);
      const float p = __expf(v - Mn);
      Lrow[r] = Lrow[r] * alpha + grpsum16(p);
      Mrow[r] = Mn;
#pragma unroll
      for (int t = 0; t < 4; ++t) oacc[t][r] *= alpha;
      pbuf[wid][r + 8 * half][l16] = p;  // C-layout row -> LDS
    }
    asm volatile("s_wait_dscnt 0x0" ::: "memory");  // wave-local LDS RAW
    // O(16x64) += P(16x16) @ Vtile(16x64)
#pragma unroll
    for (int k4 = 0; k4 < 16; k4 += 4) {
      const int kk = k4 + 2 * half;
      v2f a;  // A frag of P: row = lane%16, K = kk,kk+1
      a.x = pbuf[wid][l16][kk];
      a.y = pbuf[wid][l16][kk + 1];
#pragma unroll
      for (int t = 0; t < 4; ++t) {
        v2f bb;  // B frag of V: rows kk/kk+1, col = t*16 + lane%16
        bb.x = vp[(size_t)(kt + kk) * Dc + t * 16 + l16];
        bb.y = vp[(size_t)(kt + kk + 1) * Dc + t * 16 + l16];
        oacc[t] = __builtin_amdgcn_wmma_f32_16x16x4_f32(false, a, false, bb,
                                                        (short)0, oacc[t], false, false);
      }
    }
    asm volatile("s_wait_dscnt 0x0" ::: "memory");  // WAR before next tile's P store
  }
  // finalize: O /= rowsum, scatter back into [B,S,D] at head offset
#pragma unroll
  for (int t = 0; t < 4; ++t) {
#pragma unroll
    for (int r = 0; r < 8; ++r) {
      const int srow = q0 + r + 8 * half;
      O[(size_t)(b * Sc + srow) * Dc + h * HDc + t * 16 + l16] = oacc[t][r] / Lrow[r];
    }
  }
}

// ---------------- fused residual add + LayerNorm (one row per block) ----------------
__global__ __launch_bounds__(256)
void add_ln_kernel(const float* __restrict__ a, const float* __restrict__ xin,
                   const float* __restrict__ g, const float* __restrict__ beta,
                   float* __restrict__ out) {
  __shared__ float red[8];
  const int row = blockIdx.x;
  const int tid = threadIdx.x;
  const int wid = tid >> 5, lane = tid & 31;
  const size_t base = (size_t)row * Dc;
  const float v0 = a[base + tid] + xin[base + tid];
  const float v1 = a[base + tid + 256] + xin[base + tid + 256];
  float s = v0 + v1;
#pragma unroll
  for (int m = 1; m < 32; m <<= 1) s += __shfl_xor(s, m);
  if (lane == 0) red[wid] = s;
  __syncthreads();
  float tot = 0.0f;
#pragma unroll
  for (int i = 0; i < 8; ++i) tot += red[i];
  const float mean = tot * (1.0f / Dc);
  __syncthreads();
  const float d0 = v0 - mean, d1 = v1 - mean;
  float q = d0 * d0 + d1 * d1;
#pragma unroll
  for (int m = 1; m < 32; m <<= 1) q += __shfl_xor(q, m);
  if (lane == 0) red[wid] = q;
  __syncthreads();
  float qt = 0.0f;
#pragma unroll
  for (int i = 0; i < 8; ++i) qt += red[i];
  const float rstd = rsqrtf(qt * (1.0f / Dc) + EPSc);
  out[base + tid] = d0 * rstd * g[tid] + beta[tid];
  out[base + tid + 256] = d1 * rstd * g[tid + 256] + beta[tid + 256];
}

// ---------------- host orchestration ----------------
extern "C" void kernel_launch(void* const* d_in, const int* in_sizes, int n_in,
                              void* d_out, int out_size, void* d_ws, size_t ws_size,
                              hipStream_t stream) {
  const int* tokens = (const int*)d_in[0];
  const float* emb = (const float*)d_in[1];
  const float* wq = (const float*)d_in[2];
  const float* wk = (const float*)d_in[3];
  const float* wv = (const float*)d_in[4];
  const float* wo = (const float*)d_in[5];
  const float* w1 = (const float*)d_in[6];
  const float* bf1 = (const float*)d_in[7];
  const float* w2 = (const float*)d_in[8];
  const float* bf2 = (const float*)d_in[9];
  const float* ln1_g = (const float*)d_in[10];
  const float* ln1_b = (const float*)d_in[11];
  const float* ln2_g = (const float*)d_in[12];
  const float* ln2_b = (const float*)d_in[13];
  (void)in_sizes; (void)n_in; (void)out_size; (void)ws_size;

  float* x = (float*)d_ws;        // [B,S,D]
  float* t0 = x + BSD;            // [B,S,D] scratch (proj / ff2)
  float* pool = t0 + BSD;         // 4*BSD pool
  float* xq = pool;
  float* xk = pool + BSD;
  float* xv = pool + 2 * BSD;
  float* ob = pool + 3 * BSD;
  float* ff1 = pool;              // reuses Q/K/V/O region after attention is consumed

  const dim3 blk(256);
  embed_pe_kernel<<<BS, blk, 0, stream>>>(tokens, emb, x);

  const int gD = (BS / 64) * (Dc / 16) / 8;  // 256 blocks (N = D)
  const int gF = (BS / 64) * (Fc / 16) / 8;  // 1024 blocks (N = F)
  const int gA = (Bc * Hc * (Sc / 16)) / 8;  // 256 blocks for attention

  for (int l = 0; l < Lc; ++l) {
    const float* Wq = wq + (size_t)l * Dc * Dc;
    const float* Wk = wk + (size_t)l * Dc * Dc;
    const float* Wv = wv + (size_t)l * Dc * Dc;
    const float* Wo = wo + (size_t)l * Dc * Dc;
    const float* W1 = w1 + (size_t)l * Dc * Fc;
    const float* W2 = w2 + (size_t)l * Fc * Dc;

    gemm_wmma_f32<<<gD, blk, 0, stream>>>(x, Wq, nullptr, xq, BS, Dc, Dc, 0);
    gemm_wmma_f32<<<gD, blk, 0, stream>>>(x, Wk, nullptr, xk, BS, Dc, Dc, 0);
    gemm_wmma_f32<<<gD, blk, 0, stream>>>(x, Wv, nullptr, xv, BS, Dc, Dc, 0);
    flash_attn_wmma<<<gA, blk, 0, stream>>>(xq, xk, xv, ob);
    gemm_wmma_f32<<<gD, blk, 0, stream>>>(ob, Wo, nullptr, t0, BS, Dc, Dc, 0);
    add_ln_kernel<<<BS, blk, 0, stream>>>(t0, x, ln1_g + (size_t)l * Dc,
                                          ln1_b + (size_t)l * Dc, x);
    gemm_wmma_f32<<<gF, blk, 0, stream>>>(x, W1, bf1 + (size_t)l * Fc, ff1, BS, Fc, Dc, 1);
    gemm_wmma_f32<<<gD, blk, 0, stream>>>(ff1, W2, bf2 + (size_t)l * Dc, t0, BS, Dc, Fc, 0);
    float* dst = (l == Lc - 1) ? (float*)d_out : x;
    add_ln_kernel<<<BS, blk, 0, stream>>>(t0, x, ln2_g + (size_t)l * Dc,
                                          ln2_b + (size_t)l * Dc, dst);
  }
}